// TemporalFusion_57312043598530
// MI455X (gfx1250) — compile-verified
//
#include <hip/hip_runtime.h>

// ---------------------------------------------------------------------------
// Problem constants (from reference): x (64,256,56,56) f32, w_conv (144,9),
// b_conv (144), gamma/beta (256). Output (32,256,56,56) f32.
// ---------------------------------------------------------------------------
#define NT   64
#define N2   32
#define C    256
#define G    8
#define CG   32
#define H    56
#define W    56
#define PIX  (H * W)          // 3136
#define NQ   144              // 2*K*K*G
#define NK72 72               // K*K*G

// ws layout (bytes):
//   corr  : N2*72*PIX floats  = 7,225,344 f  @ 0
//   xcorr : N2*144*PIX floats = 14,450,688 f @ 28,901,376
//   part  : 256*32*2 doubles               @ 86,704,128
//   stats : 256*2 floats                   @ 86,835,200
#define CORR_ELEMS   ((size_t)N2 * NK72 * PIX)
#define XCORR_ELEMS  ((size_t)N2 * NQ * PIX)
#define PART_OFF_B   ((CORR_ELEMS + XCORR_ELEMS) * 4)
#define STATS_OFF_B  (PART_OFF_B + (size_t)C * N2 * 2 * 8)

typedef float v2f __attribute__((ext_vector_type(2)));
typedef float v8f __attribute__((ext_vector_type(8)));

// ---------------------------------------------------------------------------
// Kernel 1: temporal correlation.
// corr[n][ij*8+g][y*56+x] = sum_{cg<32} x[2n][g*32+cg][y][x] *
//                                       x[2n+1][g*32+cg][clamp(y+i-1)][clamp(x+j-1)]
// grid = N2*G*H blocks, 64 threads (x dim, 56 active).
// ---------------------------------------------------------------------------
__global__ void k_corr(const float* __restrict__ x, float* __restrict__ corr) {
    int b = blockIdx.x;
    int y = b % H; b /= H;
    int g = b % G; b /= G;
    int n = b;
    int xc = threadIdx.x;
    if (xc >= W) return;

    const float* xe = x + ((size_t)(2 * n) * C + g * CG) * PIX;
    const float* xo = x + ((size_t)(2 * n + 1) * C + g * CG) * PIX;

    int ym[3] = { max(y - 1, 0), y, min(y + 1, H - 1) };
    int xm[3] = { max(xc - 1, 0), xc, min(xc + 1, W - 1) };

    float acc[9] = {0.f,0.f,0.f,0.f,0.f,0.f,0.f,0.f,0.f};
    for (int cg = 0; cg < CG; ++cg) {
        float e = xe[(size_t)cg * PIX + y * W + xc];
        const float* oc = xo + (size_t)cg * PIX;
        #pragma unroll
        for (int i = 0; i < 3; ++i)
            #pragma unroll
            for (int j = 0; j < 3; ++j)
                acc[i * 3 + j] = fmaf(e, oc[ym[i] * W + xm[j]], acc[i * 3 + j]);
    }
    size_t base = (size_t)n * NK72 * PIX + y * W + xc;
    #pragma unroll
    for (int ij = 0; ij < 9; ++ij)
        corr[base + (size_t)(ij * G + g) * PIX] = acc[ij];
}

// ---------------------------------------------------------------------------
// Kernel 2: grouped 1x1 conv as block-diagonal 144x72 fp32 matmul via WMMA.
// xcorr[n][q][p] = sum_k Wfull[q][k] * corr[n][k][p] + b_conv[q]
//   Wfull[q][k] = (k/9 == q/18) ? w_conv[q][k%9] : 0
// One wave per 16x16 output tile; K = 72 in 18 chunks of 4.
// grid = 7056 blocks * 256 thr = 56448 waves = 32 n * 9 mtiles * 196 ntiles.
// All waves full -> EXEC all ones (WMMA requirement).
// ---------------------------------------------------------------------------
__device__ __forceinline__ float wfull(const float* __restrict__ w, int q, int k) {
    float v = w[q * 9 + (k % 9)];           // always in-bounds (q<144, k<72)
    return (k / 9 == q / 18) ? v : 0.f;
}

__global__ void k_einsum_wmma(const float* __restrict__ corr,
                              const float* __restrict__ w_conv,
                              const float* __restrict__ b_conv,
                              float* __restrict__ xcorr) {
    int lane  = threadIdx.x & 31;
    int hh    = lane >> 4;                  // lane half: 0 or 1
    int l16   = lane & 15;
    int gwave = blockIdx.x * (blockDim.x >> 5) + (threadIdx.x >> 5);

    int ntile = gwave % 196; gwave /= 196;
    int mt    = gwave % 9;   gwave /= 9;
    int n     = gwave;                      // 0..31

    int mb   = mt * 16;                     // output-row (q) base
    int pcol = ntile * 16 + l16;            // pixel column for B/D
    int qa   = mb + l16;                    // A row held by this lane

    const float* crn = corr + (size_t)n * NK72 * PIX;

    v8f c = {0.f,0.f,0.f,0.f,0.f,0.f,0.f,0.f};
    for (int k0 = 0; k0 < NK72; k0 += 4) {
        int ka = k0 + 2 * hh;               // (v0,lo)=K0 (v1,lo)=K1 (v0,hi)=K2 (v1,hi)=K3
        v2f a, bv;
        a.x  = wfull(w_conv, qa, ka);
        a.y  = wfull(w_conv, qa, ka + 1);
        bv.x = crn[(size_t)ka * PIX + pcol];
        bv.y = crn[(size_t)(ka + 1) * PIX + pcol];
        c = __builtin_amdgcn_wmma_f32_16x16x4_f32(
                /*neg_a=*/false, a, /*neg_b=*/false, bv,
                /*c_mod=*/(short)0, c, /*reuse_a=*/false, /*reuse_b=*/false);
    }

    // D layout: vgpr r -> row mb + r + 8*hh, col = pcol
    float* xn = xcorr + (size_t)n * NQ * PIX;
    #pragma unroll
    for (int r = 0; r < 8; ++r) {
        int q = mb + r + 8 * hh;
        xn[(size_t)q * PIX + pcol] = c[r] + b_conv[q];
    }
}

// ---------------------------------------------------------------------------
// Kernel 3: reconstruction (unnormalized output).
// out[n][cgi*8+gi][y][x] = sum_{i,j} xe[y+i-1][x+j-1]*xcorr[(ij)*16+2gi]
//                                  + xo[y+i-1][x+j-1]*xcorr[(ij)*16+2gi+1]
// grid = N2*H blocks, 64 threads.
// ---------------------------------------------------------------------------
__global__ void k_recon(const float* __restrict__ x,
                        const float* __restrict__ xcorr,
                        float* __restrict__ out) {
    int b = blockIdx.x;
    int y = b % H;
    int n = b / H;
    int xc = threadIdx.x;
    if (xc >= W) return;

    int ym[3] = { max(y - 1, 0), y, min(y + 1, H - 1) };
    int xm[3] = { max(xc - 1, 0), xc, min(xc + 1, W - 1) };
    int pix = y * W + xc;

    const float* xcn = xcorr + (size_t)n * NQ * PIX + pix;

    for (int gi = 0; gi < G; ++gi) {
        float we[9], wo[9];
        #pragma unroll
        for (int ij = 0; ij < 9; ++ij) {
            we[ij] = xcn[(size_t)(ij * 16 + 2 * gi)     * PIX];
            wo[ij] = xcn[(size_t)(ij * 16 + 2 * gi + 1) * PIX];
        }
        for (int cgi = 0; cgi < CG; ++cgi) {
            int ch = cgi * G + gi;
            const float* xe = x + ((size_t)(2 * n) * C + ch) * PIX;
            const float* xo = x + ((size_t)(2 * n + 1) * C + ch) * PIX;
            float acc = 0.f;
            #pragma unroll
            for (int i = 0; i < 3; ++i)
                #pragma unroll
                for (int j = 0; j < 3; ++j) {
                    int t = ym[i] * W + xm[j];
                    acc = fmaf(xe[t], we[i * 3 + j], acc);
                    acc = fmaf(xo[t], wo[i * 3 + j], acc);
                }
            out[((size_t)n * C + ch) * PIX + pix] = acc;
        }
    }
}

// ---------------------------------------------------------------------------
// Kernel 4: deterministic BN partial sums per (channel, frame) in double.
// grid = (C, N2), 256 threads, fixed-order tree reduction.
// ---------------------------------------------------------------------------
__global__ void k_bnpart(const float* __restrict__ out, double* __restrict__ part) {
    int ch = blockIdx.x;
    int n  = blockIdx.y;
    int tid = threadIdx.x;
    const float* p = out + ((size_t)n * C + ch) * PIX;

    double s = 0.0, s2 = 0.0;
    for (int i = tid; i < PIX; i += 256) {
        double v = (double)p[i];
        s += v; s2 += v * v;
    }
    __shared__ double ls[256], lq[256];
    ls[tid] = s; lq[tid] = s2;
    __syncthreads();
    for (int off = 128; off > 0; off >>= 1) {
        if (tid < off) { ls[tid] += ls[tid + off]; lq[tid] += lq[tid + off]; }
        __syncthreads();
    }
    if (tid == 0) {
        part[((size_t)ch * N2 + n) * 2 + 0] = ls[0];
        part[((size_t)ch * N2 + n) * 2 + 1] = lq[0];
    }
}

// ---------------------------------------------------------------------------
// Kernel 5: per-channel scale/shift.
// ---------------------------------------------------------------------------
__global__ void k_bnstats(const double* __restrict__ part,
                          const float* __restrict__ gamma,
                          const float* __restrict__ beta,
                          float* __restrict__ stats) {
    int ch = threadIdx.x;
    if (ch >= C) return;
    double s = 0.0, s2 = 0.0;
    for (int n = 0; n < N2; ++n) {
        s  += part[((size_t)ch * N2 + n) * 2 + 0];
        s2 += part[((size_t)ch * N2 + n) * 2 + 1];
    }
    const double cnt = (double)N2 * PIX;
    double mean = s / cnt;
    double var  = s2 / cnt - mean * mean;
    double inv  = 1.0 / sqrt(var + 1e-5);
    double sc   = (double)gamma[ch] * inv;
    stats[ch * 2 + 0] = (float)sc;
    stats[ch * 2 + 1] = (float)((double)beta[ch] - mean * sc);
}

// ---------------------------------------------------------------------------
// Kernel 6: in-place normalize. grid = N2*C blocks, 256 threads.
// ---------------------------------------------------------------------------
__global__ void k_bnapply(float* __restrict__ out, const float* __restrict__ stats) {
    int b  = blockIdx.x;           // n*C + ch
    int ch = b % C;
    float sc = stats[ch * 2 + 0];
    float sh = stats[ch * 2 + 1];
    float* p = out + (size_t)b * PIX;
    for (int i = threadIdx.x; i < PIX; i += 256)
        p[i] = fmaf(p[i], sc, sh);
}

// ---------------------------------------------------------------------------
extern "C" void kernel_launch(void* const* d_in, const int* in_sizes, int n_in,
                              void* d_out, int out_size, void* d_ws, size_t ws_size,
                              hipStream_t stream) {
    const float* x      = (const float*)d_in[0];
    const float* w_conv = (const float*)d_in[1];
    const float* b_conv = (const float*)d_in[2];
    const float* gamma  = (const float*)d_in[3];
    const float* beta   = (const float*)d_in[4];
    float* out = (float*)d_out;

    float*  ws_corr  = (float*)d_ws;
    float*  ws_xcorr = ws_corr + CORR_ELEMS;
    double* ws_part  = (double*)((char*)d_ws + PART_OFF_B);
    float*  ws_stats = (float*)((char*)d_ws + STATS_OFF_B);

    k_corr       <<<N2 * G * H, 64, 0, stream>>>(x, ws_corr);
    k_einsum_wmma<<<7056, 256, 0, stream>>>(ws_corr, w_conv, b_conv, ws_xcorr);
    k_recon      <<<N2 * H, 64, 0, stream>>>(x, ws_xcorr, out);
    k_bnpart     <<<dim3(C, N2), 256, 0, stream>>>(out, ws_part);
    k_bnstats    <<<1, 256, 0, stream>>>(ws_part, gamma, beta, ws_stats);
    k_bnapply    <<<N2 * C, 256, 0, stream>>>(out, ws_stats);
}